// ContrastMemory_43748536877734
// MI455X (gfx1250) — compile-verified
//
#include <hip/hip_runtime.h>
#include <math.h>

typedef float v2f __attribute__((ext_vector_type(2)));
typedef float v8f __attribute__((ext_vector_type(8)));

#define N_OUT  100000
#define DDIM   128
#define BATCH  128
#define KP1    4097
#define NTILE  257            // ceil(4097/16)
#define INV_T  (1.0f/0.07f)
#define MOM    0.5f

// ---------------------------------------------------------------------------
// Pass 1: gathered dot products via V_WMMA_F32_16X16X4_F32 + exp, and
// deterministic per-(side,b) partial sums for Z.
// grid = (BATCH, 2), block = 256 (8 waves). side 0: out_v1 = exp(<mem2,v1>/T)
// ---------------------------------------------------------------------------
__global__ __launch_bounds__(256) void dot_exp_kernel(
    const float* __restrict__ v1, const float* __restrict__ v2,
    const int*   __restrict__ idx,
    const float* __restrict__ mem1, const float* __restrict__ mem2,
    float* __restrict__ outbase, float* __restrict__ partials)
{
    __shared__ float vb[DDIM];
    __shared__ float wsum[8];

    const int b    = blockIdx.x;
    const int side = blockIdx.y;
    const float* mem  = side ? mem1 : mem2;
    const float* feat = (side ? v2 : v1) + b * DDIM;
    float* out = outbase + side * (BATCH * KP1) + b * KP1;

    const int tid = threadIdx.x;
    if (tid < DDIM) vb[tid] = feat[tid];
    __syncthreads();

    const int wave = tid >> 5;
    const int lane = tid & 31;
    const int ml   = lane & 15;   // row within 16-row tile
    const int hi   = lane >> 4;   // 0: K={0,1}, 1: K={2,3} (A); same for B rows
    const int doff = hi << 1;

    float lsum = 0.0f;

    for (int t = wave; t < NTILE; t += 8) {          // uniform trip count per wave
        const int kbase  = t << 4;
        const int k      = kbase + ml;
        const int kvalid = (k < KP1);
        const int row    = kvalid ? idx[b * KP1 + k] : 0;
        const float* rowp = mem + (long long)row * DDIM + doff;

        v8f c = {0.f,0.f,0.f,0.f,0.f,0.f,0.f,0.f};
        #pragma unroll
        for (int d = 0; d < DDIM; d += 4) {
            v2f a  = *(const v2f*)(rowp + d);          // A: 16x4 fp32 tile chunk
            v2f bv = *(const v2f*)(&vb[d + doff]);     // B: v[b] chunk bcast over N
            c = __builtin_amdgcn_wmma_f32_16x16x4_f32(
                    false, a, false, bv, (short)0, c, false, false);
        }

        // C layout: lane l, element j -> M = j + 8*hi, N = l%16 (all N identical)
        float tsum = 0.0f;
        #pragma unroll
        for (int j = 0; j < 8; ++j) {
            const int kk = kbase + (hi << 3) + j;
            const float e = __expf(c[j] * INV_T);
            if (kk < KP1) {
                tsum += e;
                if (ml == 0) out[kk] = e;              // lanes 0 & 16 own M 0-7 / 8-15
            }
        }
        if (ml == 0) lsum += tsum;
    }

    // combine lanes 0 and 16, then fixed-order block reduce
    lsum += __shfl_xor(lsum, 16);
    if (lane == 0) wsum[wave] = lsum;
    __syncthreads();
    if (tid == 0) {
        float p = 0.0f;
        for (int w = 0; w < 8; ++w) p += wsum[w];
        partials[side * BATCH + b] = p;
    }
}

// ---------------------------------------------------------------------------
// Pass 2: fixed-order sum of the 128 per-b partials -> 1/Z per side
// ---------------------------------------------------------------------------
__global__ void finalize_scales_kernel(float* __restrict__ ws)
{
    const int s = threadIdx.x;
    if (s < 2) {
        float acc = 0.0f;
        for (int i = 0; i < BATCH; ++i) acc += ws[s * BATCH + i];
        // Z = (sum / (B*(K+1))) * N  ->  scale = B*(K+1) / (sum * N)
        ws[2 * BATCH + s] = ((float)BATCH * (float)KP1) / (acc * (float)N_OUT);
    }
}

// ---------------------------------------------------------------------------
// Pass 3: normalize the exp outputs by 1/Z
// ---------------------------------------------------------------------------
__global__ __launch_bounds__(256) void scale_kernel(
    float* __restrict__ out, const float* __restrict__ scales)
{
    const int i = blockIdx.x * 256 + threadIdx.x;
    if (i < 2 * BATCH * KP1) {
        out[i] *= scales[i >= BATCH * KP1];
    }
}

// ---------------------------------------------------------------------------
// Memory-bank copy (float4 streaming)
// ---------------------------------------------------------------------------
__global__ __launch_bounds__(256) void copy_mem_kernel(
    const float4* __restrict__ m1, const float4* __restrict__ m2,
    float4* __restrict__ d1, float4* __restrict__ d2)
{
    const long long i   = (long long)blockIdx.x * 256 + threadIdx.x;
    const long long per = (long long)N_OUT * DDIM / 4;
    if (i < per)            d1[i]       = m1[i];
    else if (i < 2 * per)   d2[i - per] = m2[i - per];
}

// ---------------------------------------------------------------------------
// Momentum blend + L2-normalize rows y[b]; grid (BATCH,2), block 128
// ---------------------------------------------------------------------------
__global__ __launch_bounds__(128) void update_rows_kernel(
    const float* __restrict__ v1, const float* __restrict__ v2,
    const int*   __restrict__ y,
    const float* __restrict__ mem1, const float* __restrict__ mem2,
    float* __restrict__ dst1, float* __restrict__ dst2)
{
    __shared__ float red[DDIM];
    const int b     = blockIdx.x;
    const int which = blockIdx.y;            // 0: (mem1, v1) -> new_memory_v1
    const int t     = threadIdx.x;
    const int row   = y[b];

    const float* mem = which ? mem2 : mem1;
    const float* v   = (which ? v2 : v1) + b * DDIM;
    float* dst       = which ? dst2 : dst1;

    const float u = MOM * mem[(long long)row * DDIM + t] + (1.0f - MOM) * v[t];
    red[t] = u * u;
    __syncthreads();
    for (int s = 64; s > 0; s >>= 1) {
        if (t < s) red[t] += red[t + s];
        __syncthreads();
    }
    const float inv = 1.0f / sqrtf(red[0]);
    dst[(long long)row * DDIM + t] = u * inv;
}

// ---------------------------------------------------------------------------
extern "C" void kernel_launch(void* const* d_in, const int* in_sizes, int n_in,
                              void* d_out, int out_size, void* d_ws, size_t ws_size,
                              hipStream_t stream)
{
    (void)in_sizes; (void)n_in; (void)out_size; (void)ws_size;

    const float* v1   = (const float*)d_in[0];
    const float* v2   = (const float*)d_in[1];
    const int*   idx  = (const int*)  d_in[2];
    const int*   y    = (const int*)  d_in[3];
    const float* mem1 = (const float*)d_in[4];
    const float* mem2 = (const float*)d_in[5];

    float* out = (float*)d_out;
    float* ws  = (float*)d_ws;   // [0..255] per-(side,b) partials, [256..257] scales

    float* new_mem1 = out + 2 * BATCH * KP1;
    float* new_mem2 = new_mem1 + (long long)N_OUT * DDIM;

    dim3 g1(BATCH, 2);
    dot_exp_kernel<<<g1, 256, 0, stream>>>(v1, v2, idx, mem1, mem2, out, ws);

    finalize_scales_kernel<<<1, 32, 0, stream>>>(ws);

    const int totalScale = 2 * BATCH * KP1;
    scale_kernel<<<(totalScale + 255) / 256, 256, 0, stream>>>(out, ws + 2 * BATCH);

    const long long totCopy = 2LL * N_OUT * DDIM / 4;
    copy_mem_kernel<<<(int)((totCopy + 255) / 256), 256, 0, stream>>>(
        (const float4*)mem1, (const float4*)mem2, (float4*)new_mem1, (float4*)new_mem2);

    dim3 g2(BATCH, 2);
    update_rows_kernel<<<g2, 128, 0, stream>>>(v1, v2, y, mem1, mem2, new_mem1, new_mem2);
}